// AttentionHead_80221399155430
// MI455X (gfx1250) — compile-verified
//
#include <hip/hip_runtime.h>
#include <math.h>

typedef __attribute__((ext_vector_type(2))) float v2f;
typedef __attribute__((ext_vector_type(8))) float v8f;

#define EMB   768
#define HS    64
#define SEQ   2048
#define BATCH 8
#define XPAD  772   // 768 + 4 : 16 A-operand rows hit 16 distinct LDS banks
#define VPAD  72    // 64 + 8  : half-wave bank sets disjoint for V B-operand reads

__device__ __forceinline__ v8f wmma_f32(v2f a, v2f b, v8f c) {
  // V_WMMA_F32_16X16X4_F32 : D = A(16x4) * B(4x16) + C(16x16), fp32 throughout
  return __builtin_amdgcn_wmma_f32_16x16x4_f32(
      /*neg_a=*/false, a, /*neg_b=*/false, b,
      /*c_mod=*/(short)0, c, /*reuse_a=*/false, /*reuse_b=*/false);
}

// ---------------------------------------------------------------------------
// Kernel 1: projections.  grid = (M/16, 3), block = 128 (4 waves).
// blockIdx.y: 0 -> Q (Wq), 1 -> K (Wk, stored transposed Kt[b][h][s]), 2 -> V.
// Each wave computes a 16x16 tile of the [16 x 64] output slab.
// ---------------------------------------------------------------------------
__global__ __launch_bounds__(128) void proj_kernel(
    const float* __restrict__ x,
    const float* __restrict__ Wq, const float* __restrict__ Wk,
    const float* __restrict__ Wv,
    float* __restrict__ Q, float* __restrict__ Kt, float* __restrict__ V) {
  __shared__ float xs[16 * XPAD];

  const int tileM = blockIdx.x;      // 0..1023 over B*S/16
  const int which = blockIdx.y;      // 0=Q 1=K 2=V
  const int tid   = threadIdx.x;
  const int lane  = tid & 31;
  const int wave  = tid >> 5;        // 0..3
  const int l16   = lane & 15;
  const int half  = lane >> 4;       // 0 or 1
  const int halfk = half * 2;        // K-pair select for A/B operands
  const int rowBase = tileM * 16;

  // stage X tile [16 x 768] -> LDS (coalesced)
  for (int i = tid; i < 16 * EMB; i += 128) {
    const int r = i / EMB, cc = i - r * EMB;
    xs[r * XPAD + cc] = x[(rowBase + r) * EMB + cc];
  }
  __syncthreads();

  const float* W = (which == 0) ? Wq : (which == 1) ? Wk : Wv;
  const int n = wave * 16 + l16;     // output column (head dim)

  v8f c = {};
  for (int kc = 0; kc < EMB / 4; ++kc) {
    const int k0 = kc * 4 + halfk;
    v2f a, b;
    a.x = xs[l16 * XPAD + k0];       // A[m=l16][k0], A[m][k0+1]
    a.y = xs[l16 * XPAD + k0 + 1];
    b.x = W[k0 * HS + n];            // B[k0][n], B[k0+1][n]
    b.y = W[(k0 + 1) * HS + n];
    c = wmma_f32(a, b, c);
  }

  if (which == 1) {
    // store K transposed: Kt[b][h][s]   (tile never crosses batch: 2048 % 16 == 0)
    const int bidx = rowBase / SEQ;
    const int s0   = rowBase - bidx * SEQ;
#pragma unroll
    for (int v = 0; v < 8; ++v) {
      const int srow = s0 + v + half * 8;
      Kt[(bidx * HS + n) * SEQ + srow] = c[v];
    }
  } else {
    float* O = (which == 0) ? Q : V;
#pragma unroll
    for (int v = 0; v < 8; ++v) {
      const int m = rowBase + v + half * 8;
      O[m * HS + n] = c[v];
    }
  }
}

// ---------------------------------------------------------------------------
// Kernel 2: flash attention.  block = 256 (8 waves); the 8 waves own 8
// consecutive 16-query tiles of the SAME batch, so K/V tiles are staged into
// LDS once per 16-key step (8x traffic reduction), double-buffered so the
// next tile's global loads overlap the current tile's 32 WMMAs.
// ---------------------------------------------------------------------------
__global__ __launch_bounds__(256) void attn_kernel(
    const float* __restrict__ Q, const float* __restrict__ Kt,
    const float* __restrict__ V, float* __restrict__ out) {
  __shared__ float Kbuf[2][HS * 16];      // [h][key_local], stride 16
  __shared__ float Vbuf[2][16 * VPAD];    // [key_local][h], stride 72
  __shared__ float plds[8][16 * 17];      // per-wave P tile, stride 17

  const int tid   = threadIdx.x;
  const int lane  = tid & 31;
  const int wave  = tid >> 5;                    // 0..7
  const int gt    = blockIdx.x * 8 + wave;       // global q-tile id
  const int b     = gt >> 7;                     // uniform across the block
  const int qbase = (gt & 127) * 16;
  const int l16   = lane & 15;
  const int half  = lane >> 4;
  const int halfk = half * 2;

  // preload Q A-operands for all 16 K-chunks of the head dim (16x64 tile)
  v2f qa[16];
  const int mrow = (b * SEQ + qbase + l16) * HS;
#pragma unroll
  for (int kc = 0; kc < 16; ++kc) {
    qa[kc].x = Q[mrow + kc * 4 + halfk];
    qa[kc].y = Q[mrow + kc * 4 + halfk + 1];
  }

  v8f acc0 = {}, acc1 = {}, acc2 = {}, acc3 = {};
  float mst[8], lst[8];
#pragma unroll
  for (int v = 0; v < 8; ++v) { mst[v] = -INFINITY; lst[v] = 0.0f; }

  float* pl = &plds[wave][0];
  const float scale = 0.125f;  // 1/sqrt(64)

  // cooperative K/V tile staging (coalesced: 16- and 64-float rows)
  auto stage = [&](int kt, int bufi) {
    float* kb = &Kbuf[bufi][0];
    float* vb = &Vbuf[bufi][0];
    const int kBase = (b * HS) * SEQ + kt * 16;        // Kt[b][h=0][kt*16]
    const int vBase = (b * SEQ + kt * 16) * HS;        // V[b][kt*16][0]
#pragma unroll
    for (int i = tid; i < HS * 16; i += 256) {         // K: [64][16]
      const int h = i >> 4, kl = i & 15;
      kb[i] = Kt[kBase + h * SEQ + kl];
    }
#pragma unroll
    for (int i = tid; i < 16 * HS; i += 256) {         // V: [16][64] -> stride 72
      const int kl = i >> 6, h = i & 63;
      vb[kl * VPAD + h] = V[vBase + kl * HS + h];
    }
  };

  stage(0, 0);

  for (int kt = 0; kt < SEQ / 16; ++kt) {
    __syncthreads();                       // staged tile for kt is visible
    const int bufi = kt & 1;
    if (kt + 1 < SEQ / 16) stage(kt + 1, bufi ^ 1);   // overlap next tile's loads

    const float* kb = &Kbuf[bufi][0];
    const float* vb = &Vbuf[bufi][0];

    // ---- scores: S(16q x 16k) = Q * K^T, via 16 fp32 WMMAs (B from LDS)
    v8f s = {};
#pragma unroll
    for (int kc = 0; kc < 16; ++kc) {
      const int h = kc * 4 + halfk;
      v2f kv;
      kv.x = kb[h * 16 + l16];             // B[h][key] = K[key][h]
      kv.y = kb[(h + 1) * 16 + l16];
      s = wmma_f32(qa[kc], kv, s);
    }

    // ---- online softmax (row = v + 8*half; columns live in the 16-lane half)
#pragma unroll
    for (int v = 0; v < 8; ++v) {
      float sv = s[v] * scale;
      float tm = sv;
      tm = fmaxf(tm, __shfl_xor(tm, 1));
      tm = fmaxf(tm, __shfl_xor(tm, 2));
      tm = fmaxf(tm, __shfl_xor(tm, 4));
      tm = fmaxf(tm, __shfl_xor(tm, 8));
      const float nm    = fmaxf(mst[v], tm);
      const float alpha = __expf(mst[v] - nm);
      const float p     = __expf(sv - nm);
      float rs = p;
      rs += __shfl_xor(rs, 1);
      rs += __shfl_xor(rs, 2);
      rs += __shfl_xor(rs, 4);
      rs += __shfl_xor(rs, 8);
      lst[v] = lst[v] * alpha + rs;
      mst[v] = nm;
      acc0[v] *= alpha; acc1[v] *= alpha; acc2[v] *= alpha; acc3[v] *= alpha;
      // D-layout -> LDS (row-major 16x16, stride 17)
      pl[(v + half * 8) * 17 + l16] = p;
    }
    // same-wave LDS ops are in-order (DScnt); no barrier needed before reads

    // ---- PV: acc(16q x 64h) += P(16x16) * V(16k x 64h), 16 fp32 WMMAs
#pragma unroll
    for (int kc = 0; kc < 4; ++kc) {
      v2f a;
      a.x = pl[l16 * 17 + kc * 4 + halfk];          // A[q=l16][klocal]
      a.y = pl[l16 * 17 + kc * 4 + halfk + 1];
      const int r0 = (kc * 4 + halfk) * VPAD;       // V rows klocal, klocal+1
      const int r1 = r0 + VPAD;
      { v2f bv; bv.x = vb[r0 + l16];      bv.y = vb[r1 + l16];      acc0 = wmma_f32(a, bv, acc0); }
      { v2f bv; bv.x = vb[r0 + 16 + l16]; bv.y = vb[r1 + 16 + l16]; acc1 = wmma_f32(a, bv, acc1); }
      { v2f bv; bv.x = vb[r0 + 32 + l16]; bv.y = vb[r1 + 32 + l16]; acc2 = wmma_f32(a, bv, acc2); }
      { v2f bv; bv.x = vb[r0 + 48 + l16]; bv.y = vb[r1 + 48 + l16]; acc3 = wmma_f32(a, bv, acc3); }
    }
  }

  // ---- finalize: out = acc / l
#pragma unroll
  for (int v = 0; v < 8; ++v) {
    const float inv = 1.0f / lst[v];
    const int orow = (b * SEQ + qbase + v + half * 8) * HS;
    out[orow + l16]      = acc0[v] * inv;
    out[orow + 16 + l16] = acc1[v] * inv;
    out[orow + 32 + l16] = acc2[v] * inv;
    out[orow + 48 + l16] = acc3[v] * inv;
  }
}

// ---------------------------------------------------------------------------
extern "C" void kernel_launch(void* const* d_in, const int* in_sizes, int n_in,
                              void* d_out, int out_size, void* d_ws, size_t ws_size,
                              hipStream_t stream) {
  (void)in_sizes; (void)n_in; (void)out_size; (void)ws_size;
  const float* x  = (const float*)d_in[0];
  const float* Wk = (const float*)d_in[1];
  const float* Wq = (const float*)d_in[2];
  const float* Wv = (const float*)d_in[3];

  float* ws = (float*)d_ws;
  const size_t plane = (size_t)BATCH * SEQ * HS;   // 1,048,576 floats
  float* Q  = ws;
  float* Kt = ws + plane;
  float* V  = ws + 2 * plane;
  float* out = (float*)d_out;

  dim3 g1(BATCH * SEQ / 16, 3);                    // 1024 x 3 tiles
  proj_kernel<<<g1, 128, 0, stream>>>(x, Wq, Wk, Wv, Q, Kt, V);

  attn_kernel<<<(BATCH * SEQ / 16) / 8, 256, 0, stream>>>(Q, Kt, V, out);
}